// Conv2d_Local_57853209477596
// MI455X (gfx1250) — compile-verified
//
#include <hip/hip_runtime.h>
#include <hip/hip_bf16.h>

typedef __attribute__((ext_vector_type(2))) float v2f;
typedef __attribute__((ext_vector_type(4))) float v4f;
typedef __attribute__((ext_vector_type(8))) float v8f;
typedef __attribute__((ext_vector_type(4))) int   v4i;

#define NFILT  128
#define NLOC   100
#define KTOT   784
#define IW     280
#define KDIM   28            // kernel size == stride (non-overlapping patches)
#define WS_    10
#define FT     32            // filters per block tile
#define LSTR   (KTOT + 4)    // 788: LDS row stride; 788%64=20 -> conflict-free b64 reads

#if __has_builtin(__builtin_amdgcn_global_load_async_to_lds_b128) && \
    __has_builtin(__builtin_amdgcn_s_wait_asynccnt)
#define USE_ASYNC 1
// Signature (from hipcc diagnostic): arg0 = v4i in AS1 (global, non-const),
// arg1 = v4i in AS3 (LDS), arg2/arg3 = immediate i32 (offset, cpol).
typedef __attribute__((address_space(1))) v4i* gv4p;
typedef __attribute__((address_space(3))) v4i* lv4p;
#else
#define USE_ASYNC 0
#endif

// Block = one (location, 32-filter tile): 4 waves x (64 batches x 32 filters).
// Weights staged to LDS once per block; fp32 WMMA 16x16x4 over K=784.
__global__ __launch_bounds__(128) void lc_conv_wmma2(
    const float* __restrict__ x,     // [256, 1, 280, 280]
    const float* __restrict__ w,     // [12800, 784]
    const float* __restrict__ bias,  // [12800, 1]
    float* __restrict__ out)         // [256, 128, 10, 10]
{
    __shared__ float lw[FT * LSTR];  // 100,864 B of the 320 KB CDNA5 LDS

    const int tid  = threadIdx.x;
    const int lane = tid & 31;
    const int wv   = tid >> 5;

    const int ftile = blockIdx.x & 3;            // 4 filter tiles of 32
    const int loc   = blockIdx.x >> 2;           // 100 locations
    const int ly = loc / WS_;
    const int lx = loc - ly * WS_;
    const int n0 = ftile * FT;

    // ---- stage weight tile [32 x 784] into LDS (row stride 788) ----
    // 32*196 = 6272 float4 transfers / 128 threads = exactly 49 per thread.
    for (int idx = tid; idx < FT * (KTOT / 4); idx += 128) {
        const int f = idx / (KTOT / 4);
        const int c = (idx - f * (KTOT / 4)) * 4;
        const float* src = w + ((size_t)(n0 + f) * NLOC + loc) * KTOT + c;
        float* dst = &lw[f * LSTR + c];
#if USE_ASYNC
        __builtin_amdgcn_global_load_async_to_lds_b128((gv4p)src, (lv4p)dst, 0, 0);
#else
        *(v4f*)dst = *(const v4f*)src;
#endif
    }
#if USE_ASYNC
    __builtin_amdgcn_s_wait_asynccnt(0);
#endif
    __syncthreads();

    const int half = lane >> 4;                  // 0: K,K+1  1: K+2,K+3
    const int lidx = lane & 15;                  // M (A) / N (B)
    const int m0   = wv * 64;                    // 64 batches per wave

    // A fragments: 4 batch sub-tiles, global loads (L2-resident stream)
    const float* abase = x + (size_t)(m0 + lidx) * (IW * IW)
                           + (size_t)ly * (KDIM * IW) + lx * KDIM + 2 * half;
    // B fragments from LDS
    const float* lb0 = &lw[lidx * LSTR + 2 * half];
    const float* lb1 = &lw[(lidx + 16) * LSTR + 2 * half];

    v8f acc[4][2] = {};

    for (int kh = 0; kh < KDIM; ++kh) {
        const float* ar  = abase + kh * IW;
        const float* br0 = lb0 + kh * KDIM;
        const float* br1 = lb1 + kh * KDIM;
        __builtin_prefetch(ar + IW, 0, 0);       // global_prefetch_b8, next x row
#pragma unroll
        for (int kw = 0; kw < KDIM; kw += 4) {   // pairs never cross a row (28%4==0)
            v2f b0 = *(const v2f*)(br0 + kw);    // ds_load_b64, conflict-free
            v2f b1 = *(const v2f*)(br1 + kw);
#pragma unroll
            for (int mi = 0; mi < 4; ++mi) {
                v2f a = *(const v2f*)(ar + (size_t)mi * (16 * IW * IW) + kw);
                acc[mi][0] = __builtin_amdgcn_wmma_f32_16x16x4_f32(
                    false, a, false, b0, (short)0, acc[mi][0], false, false);
                acc[mi][1] = __builtin_amdgcn_wmma_f32_16x16x4_f32(
                    false, a, false, b1, (short)0, acc[mi][1], false, false);
            }
        }
    }

    // C/D layout: N = lidx for all lanes; VGPR i -> M = i + 8*half.
    const float b0v = bias[(n0 + lidx) * NLOC + loc];
    const float b1v = bias[(n0 + 16 + lidx) * NLOC + loc];

#pragma unroll
    for (int mi = 0; mi < 4; ++mi) {
#pragma unroll
        for (int i = 0; i < 8; ++i) {
            const int m = m0 + mi * 16 + i + 8 * half;
            out[((size_t)m * NFILT + (n0 + lidx)) * NLOC + loc]      = acc[mi][0][i] + b0v;
            out[((size_t)m * NFILT + (n0 + 16 + lidx)) * NLOC + loc] = acc[mi][1][i] + b1v;
        }
    }
}

extern "C" void kernel_launch(void* const* d_in, const int* in_sizes, int n_in,
                              void* d_out, int out_size, void* d_ws, size_t ws_size,
                              hipStream_t stream) {
    (void)in_sizes; (void)n_in; (void)out_size; (void)d_ws; (void)ws_size;
    const float* x    = (const float*)d_in[0];
    const float* wgt  = (const float*)d_in[1];
    const float* bias = (const float*)d_in[2];
    float* out        = (float*)d_out;

    // 100 locations x 4 filter-tiles = 400 blocks of 128 threads (4 waves).
    lc_conv_wmma2<<<400, 128, 0, stream>>>(x, wgt, bias, out);
}